// DNCM_45406394253462
// MI455X (gfx1250) — compile-verified
//
#include <hip/hip_runtime.h>

typedef __attribute__((ext_vector_type(2))) float v2f;
typedef __attribute__((ext_vector_type(8))) float v8f;

#define PLANE_ELEMS 262144   // 512*512 fp32 per (batch,channel) plane
#define NPLANES     48       // 16 batches * 3 channels
#define BPP         32       // blocks per plane (== wave32 so stage-2 is one wave-reduce per plane)
#define CHUNK_F4    (PLANE_ELEMS / BPP / 4)   // 2048 float4 per block

// ---------------------------------------------------------------------------
// Stage 1: streaming per-plane partial sums. Pure bandwidth kernel:
// coalesced float4 (global_load_b128) reads, wave32 shfl_xor tree, LDS
// cross-wave combine. Deterministic (fixed reduction tree, no atomics).
// ---------------------------------------------------------------------------
__global__ __launch_bounds__(256)
void plane_sum_kernel(const float* __restrict__ I, float* __restrict__ partials) {
    const int blk   = blockIdx.x;         // 0 .. NPLANES*BPP-1
    const int plane = blk / BPP;
    const int sub   = blk % BPP;
    const float4* base =
        (const float4*)(I + (size_t)plane * PLANE_ELEMS) + (size_t)sub * CHUNK_F4;

    float s = 0.0f;
#pragma unroll
    for (int i = 0; i < CHUNK_F4 / 256; ++i) {
        float4 v = base[threadIdx.x + i * 256];   // 4 KB contiguous per iter per block
        s += (v.x + v.y) + (v.z + v.w);
    }
    // wave32 reduction
    for (int off = 16; off > 0; off >>= 1) s += __shfl_xor(s, off, 32);

    __shared__ float red[8];
    const int lid = threadIdx.x & 31;
    const int wid = threadIdx.x >> 5;
    if (lid == 0) red[wid] = s;
    __syncthreads();
    if (threadIdx.x < 32) {
        float v = (threadIdx.x < 8) ? red[threadIdx.x] : 0.0f;
        for (int off = 4; off > 0; off >>= 1) v += __shfl_xor(v, off, 32);
        if (threadIdx.x == 0) partials[blk] = v;
    }
}

// ---------------------------------------------------------------------------
// Stage 2: single wave32. Finish plane sums S[b][c], compute q = Q*1_3,
// A = (S|0) x (P|0) via one exact-f32 V_WMMA_F32_16X16X4_F32 (K=3 padded to 4),
// then out[b] = (A[b,:] . (Tm[b] @ q)) / (HW*3) using the WMMA C/D lane layout.
// ---------------------------------------------------------------------------
__global__ __launch_bounds__(32)
void finalize_kernel(const float* __restrict__ partials,
                     const float* __restrict__ T,   // (16,16,16) row-major
                     const float* __restrict__ P,   // (3,16)
                     const float* __restrict__ Q,   // (16,3)
                     float* __restrict__ out) {
    const int lane = threadIdx.x;
    __shared__ float Sbc[NPLANES];
    __shared__ float qsh[16];

    // Deterministic stage-2 reduce: BPP==32 partials per plane -> one wave tree each.
    for (int bc = 0; bc < NPLANES; ++bc) {
        float v = partials[bc * BPP + lane];
        for (int off = 16; off > 0; off >>= 1) v += __shfl_xor(v, off, 32);
        if (lane == 0) Sbc[bc] = v;
    }
    if (lane < 16) qsh[lane] = Q[lane * 3 + 0] + Q[lane * 3 + 1] + Q[lane * 3 + 2];
    __syncthreads();

    const int  i  = lane & 15;
    const bool hi = lane >= 16;

    // A operand (16x4 f32): lanes 0-15 hold K=0/1, lanes 16-31 hold K=2/3; K=3 padded 0.
    v2f a;
    a.x = hi ? Sbc[i * 3 + 2] : Sbc[i * 3 + 0];
    a.y = hi ? 0.0f           : Sbc[i * 3 + 1];
    // B operand (4x16 f32): lanes 0-15 rows K=0/1, lanes 16-31 rows K=2/3; row 3 padded 0.
    v2f b;
    b.x = hi ? P[2 * 16 + i] : P[0 * 16 + i];
    b.y = hi ? 0.0f          : P[1 * 16 + i];
    v8f c = {0.f, 0.f, 0.f, 0.f, 0.f, 0.f, 0.f, 0.f};

    // D[b][i] = sum_c S[b][c]*P[c][i]  -- exact f32 matrix op.
    v8f A = __builtin_amdgcn_wmma_f32_16x16x4_f32(false, a, false, b,
                                                  (short)0, c, false, false);

    float qreg[16];
#pragma unroll
    for (int k = 0; k < 16; ++k) qreg[k] = qsh[k];

    // C/D layout: VGPR r, lanes 0-15 -> D[r][lane], lanes 16-31 -> D[r+8][lane-16].
    const int b0 = hi ? 8 : 0;
    float res[8];
#pragma unroll
    for (int r = 0; r < 8; ++r) {
        const float* Trow = T + (size_t)(b0 + r) * 256 + (size_t)i * 16;
        float y = 0.0f;
#pragma unroll
        for (int k = 0; k < 16; ++k) y += Trow[k] * qreg[k];   // Y[b][i] = Tm[b] @ q
        float p = A[r] * y;
        // reduce over i: xor offsets 8,4,2,1 stay within each 16-lane half
        for (int off = 8; off > 0; off >>= 1) p += __shfl_xor(p, off, 32);
        res[r] = p;
    }

    const float inv = 1.0f / 786432.0f;   // 1/(512*512*3)
    if (i == 0) {
#pragma unroll
        for (int r = 0; r < 8; ++r) out[b0 + r] = res[r] * inv;
    }
}

// ---------------------------------------------------------------------------
extern "C" void kernel_launch(void* const* d_in, const int* in_sizes, int n_in,
                              void* d_out, int out_size, void* d_ws, size_t ws_size,
                              hipStream_t stream) {
    const float* I = (const float*)d_in[0];   // (16,3,512,512)
    const float* T = (const float*)d_in[1];   // (16,256)
    const float* P = (const float*)d_in[2];   // (3,16)
    const float* Q = (const float*)d_in[3];   // (16,3)
    float* out      = (float*)d_out;          // (16,1)
    float* partials = (float*)d_ws;           // NPLANES*BPP = 1536 floats (6 KB)

    plane_sum_kernel<<<dim3(NPLANES * BPP), dim3(256), 0, stream>>>(I, partials);
    finalize_kernel<<<dim3(1), dim3(32), 0, stream>>>(partials, T, P, Q, out);
}